// NanoDetLoss_37022618092244
// MI455X (gfx1250) — compile-verified
//
#include <hip/hip_runtime.h>
#include <math.h>

#define NUM_CLASSES 80
#define REG_BINS 8            // REG_MAX + 1
#define WAVES_PER_BLOCK 8     // 256 threads, wave32

typedef __attribute__((ext_vector_type(2))) float v2f;
typedef __attribute__((ext_vector_type(8))) float v8f;

// ---------- wave-level reductions (wave32) ----------
__device__ __forceinline__ float wave_sum(float v) {
#pragma unroll
  for (int o = 16; o > 0; o >>= 1) v += __shfl_xor(v, o, 32);
  return v;
}
__device__ __forceinline__ float wave_max(float v) {
#pragma unroll
  for (int o = 16; o > 0; o >>= 1) v = fmaxf(v, __shfl_xor(v, o, 32));
  return v;
}
// group-of-8 reductions (one bbox corner distribution per 8 lanes)
__device__ __forceinline__ float g8_sum(float v) {
#pragma unroll
  for (int o = 4; o > 0; o >>= 1) v += __shfl_xor(v, o, 8);
  return v;
}
__device__ __forceinline__ float g8_max(float v) {
#pragma unroll
  for (int o = 4; o > 0; o >>= 1) v = fmaxf(v, __shfl_xor(v, o, 8));
  return v;
}

__device__ __forceinline__ float sigmoidf_fast(float x) {
  return 1.0f / (1.0f + __expf(-x));
}

// One wave per anchor, grid-strided. Writes 5 partial sums per block
// (padded to 16 floats) into ws[blockIdx.x*16 + 0..15].
__global__ __launch_bounds__(256) void nanodet_loss_main(
    const float*  __restrict__ cls,      // [N,80]
    const float*  __restrict__ bbox,     // [N,32]
    const float4* __restrict__ anchors,  // [N,4]
    const float4* __restrict__ tgts,     // [N,4]
    const float*  __restrict__ lw,       // [N]
    const float*  __restrict__ strides,  // [N]
    const int*    __restrict__ labels,   // [N]
    float*        __restrict__ ws,
    int N)
{
  const int lane      = threadIdx.x & 31;
  const int waveInBlk = threadIdx.x >> 5;
  const int gwave     = blockIdx.x * WAVES_PER_BLOCK + waveInBlk;
  const int totWaves  = gridDim.x * WAVES_PER_BLOCK;

  const int g = lane >> 3;  // corner 0..3 (l,t,r,b)
  const int b = lane & 7;   // bin 0..7

  float acc_qfl = 0.f, acc_bbox = 0.f, acc_dfl = 0.f, acc_pos = 0.f, acc_wt = 0.f;

  for (int i = gwave; i < N; i += totWaves) {
    // prefetch next anchor's cls row (global_prefetch_b8)
    int nx = i + totWaves;
    if (nx < N) __builtin_prefetch(cls + (size_t)nx * NUM_CLASSES + lane, 0, 0);

    const float4 a  = anchors[i];
    const float4 t4 = tgts[i];
    const float  s  = strides[i];
    const int    lab = labels[i];
    const float  w   = lw[i];
    const bool   pos = (lab >= 0) && (lab < NUM_CLASSES);

    const float inv_s = 1.0f / s;
    const float cx = (a.x + a.z) * 0.5f * inv_s;
    const float cy = (a.y + a.w) * 0.5f * inv_s;

    // ---- integral distribution: softmax over 8 bins per corner ----
    const float p     = bbox[(size_t)i * 32 + lane];   // [4 corners][8 bins]
    const float m     = g8_max(p);
    const float e     = __expf(p - m);
    const float denom = g8_sum(e);
    const float prob  = e / denom;
    const float corner = g8_sum(prob * (float)b);      // expectation over bins
    const float logp   = p - (m + __logf(denom));      // log_softmax, per lane

    const float c0 = __shfl(corner, 0, 32);
    const float c1 = __shfl(corner, 8, 32);
    const float c2 = __shfl(corner, 16, 32);
    const float c3 = __shfl(corner, 24, 32);

    // distance2bbox (stride-normalized coords)
    const float px0 = cx - c0, py0 = cy - c1, px1 = cx + c2, py1 = cy + c3;
    const float tx0 = t4.x * inv_s, ty0 = t4.y * inv_s;
    const float tx1 = t4.z * inv_s, ty1 = t4.w * inv_s;

    // ---- aligned IoU (QFL quality target) + GIoU ----
    const float ix0 = fmaxf(px0, tx0), iy0 = fmaxf(py0, ty0);
    const float ix1 = fminf(px1, tx1), iy1 = fminf(py1, ty1);
    const float ov  = fmaxf(ix1 - ix0, 0.f) * fmaxf(iy1 - iy0, 0.f);
    const float areaA = fmaxf(px1 - px0, 0.f) * fmaxf(py1 - py0, 0.f);
    const float areaB = fmaxf(tx1 - tx0, 0.f) * fmaxf(ty1 - ty0, 0.f);
    const float uni   = areaA + areaB - ov;
    const float iou   = ov / fmaxf(uni, 1e-6f);
    const float score = pos ? iou : 0.f;

    const float uni_g = fmaxf(uni, 1e-6f);
    const float ex0 = fminf(px0, tx0), ey0 = fminf(py0, ty0);
    const float ex1 = fmaxf(px1, tx1), ey1 = fmaxf(py1, ty1);
    const float enc = fmaxf(fmaxf(ex1 - ex0, 0.f) * fmaxf(ey1 - ey0, 0.f), 1e-6f);
    const float giou = (ov / uni_g) - (enc - uni_g) / enc;

    // ---- Distribution Focal Loss: per-corner linear interp CE ----
    float d = (g == 0) ? (cx - tx0) : (g == 1) ? (cy - ty0)
            : (g == 2) ? (tx1 - cx) : (ty1 - cy);
    d = fminf(fmaxf(d, 0.f), (float)(REG_BINS - 1) - 0.01f);
    const int   dl = (int)floorf(d);
    const float wl = (float)dl + 1.f - d;
    const float wr = d - (float)dl;
    float dfl = 0.f;
    if (b == dl)     dfl += -logp * wl;
    if (b == dl + 1) dfl += -logp * wr;

    // ---- Quality Focal Loss over 80 classes (lanes stride the row) ----
    float qacc = 0.f;
    float mx   = -1e30f;
    const float* crow = cls + (size_t)i * NUM_CLASSES;
#pragma unroll
    for (int cbase = 0; cbase < NUM_CLASSES; cbase += 32) {
      const int c = cbase + lane;
      if (c < NUM_CLASSES) {
        const float x   = crow[c];
        mx = fmaxf(mx, x);
        const float sig = sigmoidf_fast(x);
        const float sp  = fmaxf(x, 0.f) + log1pf(__expf(-fabsf(x)));  // softplus
        float term;
        if (pos && c == lab) {
          const float dq = score - sig;
          term = (sp - x * score) * dq * dq;       // bce(x,score)*|score-sig|^2
        } else {
          term = sp * sig * sig;                   // bce(x,0)*sig^2
        }
        qacc += term;
      }
    }
    const float qsum = wave_sum(qacc);
    const float mxl  = wave_max(mx);
    const float wt   = pos ? sigmoidf_fast(mxl) : 0.f;  // max sigmoid prob
    const float dflsum = wave_sum(dfl);

    acc_qfl  += qsum * w;
    acc_bbox += 2.0f * (1.0f - giou) * wt;     // loss_weight=2 folded
    acc_dfl  += 0.0625f * dflsum * wt;         // 0.25/4 folded
    acc_pos  += pos ? 1.f : 0.f;
    acc_wt   += wt;
  }

  // deterministic block combine through LDS
  __shared__ float part[WAVES_PER_BLOCK][5];
  if (lane == 0) {
    part[waveInBlk][0] = acc_qfl;
    part[waveInBlk][1] = acc_bbox;
    part[waveInBlk][2] = acc_dfl;
    part[waveInBlk][3] = acc_pos;
    part[waveInBlk][4] = acc_wt;
  }
  __syncthreads();
  if (threadIdx.x < 16) {
    float ssum = 0.f;
    if (threadIdx.x < 5) {
#pragma unroll
      for (int wv = 0; wv < WAVES_PER_BLOCK; ++wv) ssum += part[wv][threadIdx.x];
    }
    ws[blockIdx.x * 16 + threadIdx.x] = ssum;   // pad cols 5..15 with zeros
  }
}

// Final reduction with V_WMMA_F32_16X16X4_F32 (exact f32 MACs).
// A[m][k] = ws[(base+k)*16 + m] (ISA 16x4 f32 A layout: lanes 0-15 hold
// K=0/1 in vgpr0/1, lanes 16-31 hold K=2/3), B = ones ->
// D[m][n] = sum_k A[m][k] + C[m][n]  => column-preserving row-chunk sums.
__global__ __launch_bounds__(32) void nanodet_loss_reduce(
    const float* __restrict__ ws, float* __restrict__ out, int nb)
{
  const int lane = threadIdx.x;
  const int half = lane >> 4;   // 0: K=0/1, 1: K=2/3
  const int m    = lane & 15;   // accumulator column (M index)

  v2f ones; ones.x = 1.f; ones.y = 1.f;
  v8f c = {};

  for (int base = 0; base < nb; base += 4) {
    v2f a;
    a.x = ws[(base + 2 * half)     * 16 + m];
    a.y = ws[(base + 2 * half + 1) * 16 + m];
    c = __builtin_amdgcn_wmma_f32_16x16x4_f32(
        /*neg_a=*/false, a, /*neg_b=*/false, ones,
        /*c_mod=*/(short)0, c, /*reuse_a=*/false, /*reuse_b=*/false);
  }

  // C/D layout: VGPR r, lanes 0-15 => M=r, N=lane; every N identical here.
  if (lane == 0) {
    const float Sq = c[0];                  // sum qfl * lw
    const float Sb = c[1];                  // 2*sum (1-giou)*wt
    const float Sd = c[2];                  // 0.0625*sum dfl*wt
    const float Sp = c[3];                  // num positives
    const float Sw = c[4];                  // sum wt
    const float num_pos = fmaxf(Sp, 1.0f);
    const float avg     = fmaxf(Sw, 1e-6f);
    out[0] = Sq / num_pos + (Sb + Sd) / avg;
  }
}

extern "C" void kernel_launch(void* const* d_in, const int* in_sizes, int n_in,
                              void* d_out, int out_size, void* d_ws, size_t ws_size,
                              hipStream_t stream) {
  (void)n_in; (void)out_size;
  const float*  cls     = (const float*)d_in[0];
  const float*  bbox    = (const float*)d_in[1];
  const float4* anchors = (const float4*)d_in[2];
  const float4* tgts    = (const float4*)d_in[3];
  const float*  lw      = (const float*)d_in[4];
  const float*  strides = (const float*)d_in[5];
  const int*    labels  = (const int*)d_in[6];
  float*        ws      = (float*)d_ws;
  float*        out     = (float*)d_out;

  const int N = in_sizes[0] / NUM_CLASSES;

  int nb = 1024;                                   // 8192 waves in flight
  int maxRows = (int)(ws_size / (16 * sizeof(float)));
  maxRows &= ~3;                                   // reduce consumes rows 4 at a time
  if (nb > maxRows) nb = maxRows;
  if (nb < 4) nb = 4;

  hipLaunchKernelGGL(nanodet_loss_main, dim3(nb), dim3(256), 0, stream,
                     cls, bbox, anchors, tgts, lw, strides, labels, ws, N);
  hipLaunchKernelGGL(nanodet_loss_reduce, dim3(1), dim3(32), 0, stream,
                     ws, out, nb);
}